// MRConv_83777632076273
// MI455X (gfx1250) — compile-verified
//
#include <hip/hip_runtime.h>

typedef __attribute__((ext_vector_type(2))) float v2f;
typedef __attribute__((ext_vector_type(8))) float v8f;

#define N_NODES 100000
#define N_EDGES 1600000
#define C_IN 128
#define C_OUT 128
#define K_TOT 256
#define N_TILES (N_NODES / 16)   // 6250 exact

// Order-preserving f32 -> u32 key so we can use native u32 atomic max.
// key(x) is strictly increasing in x over all finite/inf floats.
__device__ __forceinline__ unsigned f2key(float f) {
    unsigned u = __float_as_uint(f);
    return (u & 0x80000000u) ? ~u : (u | 0x80000000u);
}
// key 0 is the sentinel "no edge" (below every finite key) -> decode to 0.0f,
// which implements the reference's isneginf -> 0 replacement for free.
__device__ __forceinline__ float key2f(unsigned k) {
    if (k == 0u) return 0.0f;
    return __uint_as_float((k & 0x80000000u) ? (k ^ 0x80000000u) : ~k);
}

// ---------------- Phase 1: per-edge relative features, scatter-max ----------
// One wave per edge; lane handles 4 channels via float4. x and the key buffer
// both live in L2 (51.2 MB each vs 192 MB), so this is L2-atomic bound.
__global__ __launch_bounds__(256) void edge_max_kernel(
    const float* __restrict__ x, const int* __restrict__ src,
    const int* __restrict__ dst, unsigned* __restrict__ keys)
{
    long gid = (long)blockIdx.x * blockDim.x + threadIdx.x;
    int e = (int)(gid >> 5);
    if (e >= N_EDGES) return;
    int lane = (int)(gid & 31);

    int s = src[e], d = dst[e];
    const float4* xd = (const float4*)(x + (long)d * C_IN);
    const float4* xs = (const float4*)(x + (long)s * C_IN);
    float4 a = xd[lane];
    float4 b = xs[lane];

    unsigned* kp = keys + (long)d * C_IN + lane * 4;
    atomicMax(kp + 0, f2key(a.x - b.x));
    atomicMax(kp + 1, f2key(a.y - b.y));
    atomicMax(kp + 2, f2key(a.z - b.z));
    atomicMax(kp + 3, f2key(a.w - b.w));
}

// ---------------- Phase 2: fused concat-GEMM + bias + ReLU ------------------
// out = relu(x @ W[0:128] + decode(keys) @ W[128:256] + b), f32 WMMA 16x16x4.
// W (256x128 f32 = 128 KB) staged in dynamic LDS once per workgroup.
__global__ __launch_bounds__(256) void mrconv_gemm_kernel(
    const float* __restrict__ x, const unsigned* __restrict__ keys,
    const float* __restrict__ W, const float* __restrict__ bias,
    float* __restrict__ out)
{
    extern __shared__ float wlds[];   // K_TOT * C_OUT floats = 128 KB

    // Cooperative stage of the whole weight matrix into LDS.
    {
        const float4* wsrc = (const float4*)W;
        float4* wdst = (float4*)wlds;
        const int nvec = (K_TOT * C_OUT) / 4;   // 8192 float4
        for (int i = threadIdx.x; i < nvec; i += blockDim.x)
            wdst[i] = wsrc[i];
    }
    __syncthreads();

    int wave = threadIdx.x >> 5;        // wave32
    int lane = threadIdx.x & 31;
    int tile = blockIdx.x * 8 + wave;   // one 16-row tile per wave
    if (tile >= N_TILES) return;        // wave-uniform guard (EXEC stays all-1s)

    int rowBase = tile * 16;
    int m       = lane & 15;            // A: M index, B/C/D: N index
    int khalf   = (lane >> 4) << 1;     // A/B: lanes 16..31 hold K+2,K+3
    const float*    xrow = x    + (long)(rowBase + m) * C_IN;
    const unsigned* krow = keys + (long)(rowBase + m) * C_IN;

    v8f acc[8];
    #pragma unroll
    for (int t = 0; t < 8; ++t)
        #pragma unroll
        for (int j = 0; j < 8; ++j) acc[t][j] = 0.0f;

    // ---- x @ W[0:128] ----
    for (int k = 0; k < C_IN; k += 4) {
        int kb = k + khalf;
        v2f a; a.x = xrow[kb]; a.y = xrow[kb + 1];
        #pragma unroll
        for (int t = 0; t < 8; ++t) {
            int col = t * 16 + m;
            v2f b;
            b.x = wlds[kb * C_OUT + col];
            b.y = wlds[(kb + 1) * C_OUT + col];
            acc[t] = __builtin_amdgcn_wmma_f32_16x16x4_f32(
                false, a, false, b, (short)0, acc[t], false, false);
        }
    }
    // ---- max_diff @ W[128:256] (decode keys on the fly) ----
    for (int k = 0; k < C_IN; k += 4) {
        int kb = k + khalf;
        v2f a; a.x = key2f(krow[kb]); a.y = key2f(krow[kb + 1]);
        #pragma unroll
        for (int t = 0; t < 8; ++t) {
            int col = t * 16 + m;
            v2f b;
            b.x = wlds[(C_IN + kb) * C_OUT + col];
            b.y = wlds[(C_IN + kb + 1) * C_OUT + col];
            acc[t] = __builtin_amdgcn_wmma_f32_16x16x4_f32(
                false, a, false, b, (short)0, acc[t], false, false);
        }
    }

    // C/D layout: VGPR j -> rows (j, j+8) split across lane halves, N = lane&15.
    int rofs = (lane >> 4) * 8;
    #pragma unroll
    for (int t = 0; t < 8; ++t) {
        int col = t * 16 + m;
        float bv = bias[col];
        #pragma unroll
        for (int j = 0; j < 8; ++j) {
            float v = acc[t][j] + bv;
            out[(long)(rowBase + rofs + j) * C_OUT + col] = v > 0.0f ? v : 0.0f;
        }
    }
}

extern "C" void kernel_launch(void* const* d_in, const int* in_sizes, int n_in,
                              void* d_out, int out_size, void* d_ws, size_t ws_size,
                              hipStream_t stream) {
    const float* x    = (const float*)d_in[0];
    const float* W    = (const float*)d_in[1];
    const float* bias = (const float*)d_in[2];
    const int*   src  = (const int*)d_in[3];
    const int*   dst  = (const int*)d_in[4];
    float* out = (float*)d_out;
    unsigned* keys = (unsigned*)d_ws;   // N_NODES * C_IN u32 = 51.2 MB

    // Sentinel key 0 == "no incoming edge" (below every finite key).
    hipMemsetAsync(keys, 0, (size_t)N_NODES * C_IN * sizeof(unsigned), stream);

    {   // Phase 1: one wave per edge.
        long threads = (long)N_EDGES * 32;
        int blocks = (int)((threads + 255) / 256);
        edge_max_kernel<<<blocks, 256, 0, stream>>>(x, src, dst, keys);
    }
    {   // Phase 2: 8 waves/block, one 16-row tile per wave.
        int blocks = (N_TILES + 7) / 8;
        size_t lds = (size_t)K_TOT * C_OUT * sizeof(float);   // 128 KB
        mrconv_gemm_kernel<<<blocks, 256, lds, stream>>>(x, keys, W, bias, out);
    }
}